// CubicSplineUpsampling_90305982365834
// MI455X (gfx1250) — compile-verified
//
#include <hip/hip_runtime.h>

// Cubic B-spline prefilter + x2 upsampling for volume (2,3,96,96,96) -> (2,3,192,192,192), f32.
// Bandwidth-bound: ~191 MB HBM floor (~8us @ 23.3 TB/s). Intermediates kept L2-resident
// (192 MB L2), output written with non-temporal stores so it doesn't evict them.
// WMMA path: V_WMMA_F32_16X16X4_F32 computes 16 axis-outputs x 16 inner elems per 3 chained MMAs.

namespace {

constexpr float  P     = -0.26794919243112270647f;      // sqrt(3) - 2
constexpr double PD    = -0.2679491924311227064725537;  // double pole for weight generation
constexpr float  GAIN  = 6.0f;                          // (1-p)(1-1/p) == 6 exactly
constexpr float  ALAST = P / (P - 1.0f);                // p/(p-1)

// 4-tap phase weights, exact: even = {1,121,235,27}/384, odd = reverse.
// Numerators fit in bytes -> packed in one u32; select = shift/and/cvt/mul (branchless).
// For t in [0,3]: 3-t == t^3, so odd-phase reversal is XOR with 3.
__device__ __forceinline__ float wtap(int parity, int t) {
  unsigned u = (unsigned)(t ^ (parity * 3)) & 3u;
  return (float)((0x1BEB7901u >> (8u * u)) & 0xFFu) * (1.0f / 384.0f);
}

__device__ __forceinline__ int iclamp95(int v) { return v < 0 ? 0 : (v > 95 ? 95 : v); }

// ---------------------------------------------------------------------------
// Prefilter along contiguous x axis (rows of 96 floats). LDS transpose:
// 128 rows per block, row pitch 97 -> conflict-free (33*t mod 64 bijective).
// ---------------------------------------------------------------------------
__global__ __launch_bounds__(128)
void pf_x_kernel(const float* __restrict__ src, float* __restrict__ dst) {
  __shared__ float tile[128 * 97];
  const int t = threadIdx.x;
  const long long row0 = (long long)blockIdx.x * 128;
  const float* gsrc = src + row0 * 96;
  float*       gdst = dst + row0 * 96;

  #pragma unroll 8
  for (int k = 0; k < 96; ++k) {               // coalesced load of 128x96 chunk
    int idx = t + k * 128;
    int r = idx / 96, c = idx - r * 96;
    tile[r * 97 + c] = GAIN * gsrc[idx];
  }
  __syncthreads();

  float* v = &tile[t * 97];                    // this lane's row (xm = GAIN*x)
  float sum = 0.f; double pw = 1.0;
  for (int i = 0; i < 96; ++i) { sum = fmaf((float)pw, v[i], sum); pw *= PD; }
  float carry = fmaf(sum, P, v[0]);            // causal init (mirror BC; p^96 underflows)
  v[0] = carry;
  for (int i = 1; i < 96; ++i) { carry = fmaf(P, carry, v[i]); v[i] = carry; }
  carry = v[95] * ALAST;                       // anticausal init
  v[95] = carry;
  for (int i = 94; i >= 0; --i) { carry = P * (carry - v[i]); v[i] = carry; }
  __syncthreads();

  #pragma unroll 8
  for (int k = 0; k < 96; ++k) {               // coalesced store
    int idx = t + k * 128;
    int r = idx / 96, c = idx - r * 96;
    gdst[idx] = tile[r * 97 + c];
  }
}

// ---------------------------------------------------------------------------
// In-place prefilter along a strided axis (len 96, stride S == inner size).
// One row per thread; lanes walk the contiguous inner dim -> coalesced.
// ---------------------------------------------------------------------------
__global__ __launch_bounds__(256)
void pf_strided_kernel(float* __restrict__ a, int S, int nrows) {
  int rho = blockIdx.x * 256 + threadIdx.x;
  if (rho >= nrows) return;
  int o = rho / S;
  int i = rho - o * S;
  long long base = (long long)o * 96 * S + i;

  float v[96];
  #pragma unroll
  for (int k = 0; k < 96; ++k) v[k] = GAIN * a[base + (long long)k * S];

  float sum = 0.f; double pw = 1.0;
  #pragma unroll
  for (int k = 0; k < 96; ++k) { sum = fmaf((float)pw, v[k], sum); pw *= PD; }
  float carry = fmaf(sum, P, v[0]);
  v[0] = carry;
  #pragma unroll
  for (int k = 1; k < 96; ++k) { carry = fmaf(P, carry, v[k]); v[k] = carry; }
  carry = v[95] * ALAST; v[95] = carry;
  #pragma unroll
  for (int k = 94; k >= 0; --k) { carry = P * (carry - v[k]); v[k] = carry; }

  #pragma unroll
  for (int k = 0; k < 96; ++k) a[base + (long long)k * S] = v[k];
}

// ---------------------------------------------------------------------------
// WMMA x2 upsample along a strided axis. View: (Outer, 96, inner) -> (Outer, 192, inner).
// Per wave: one 16(axis-out) x 16(inner) tile. 12 source lines -> 3 K=4 chunks:
//   D = sum_g W_g(16x4) x X_g(4x16)   via V_WMMA_F32_16X16X4_F32.
// A layout (ISA 7.12.2): lanes 0-15 -> {K0,K1}, lanes 16-31 -> {K2,K3}, M = lane&15.
// B layout (analogous): lanes 0-15 -> {K0,K1}, lanes 16-31 -> {K2,K3}, N = lane&15.
// D layout: VGPR v: M = v + 8*(lane>=16), N = lane&15.
// Grids are exact multiples of full waves and the kernel is divergence-free
// (all selects are pure bit-arithmetic), so EXEC is all-ones at every WMMA.
// ---------------------------------------------------------------------------
typedef float v2f __attribute__((ext_vector_type(2)));
typedef float v8f __attribute__((ext_vector_type(8)));

__global__ __launch_bounds__(256)
void ups_axis_wmma_kernel(const float* __restrict__ in, float* __restrict__ out,
                          int inner, int tiles_inner) {
  const int NIN = 96;
  const int w    = blockIdx.x * (blockDim.x >> 5) + (threadIdx.x >> 5);
  const int lane = threadIdx.x & 31;
  const int hi = lane >> 4, lo = lane & 15;

  int ti  = w % tiles_inner;
  int tmp = w / tiles_inner;
  int tz  = tmp % 12;               // 192/16 tiles along upsampled axis
  int o   = tmp / 12;

  const int zo0 = tz * 16;
  const int x0  = ti * 16;
  const int base_src = (zo0 >> 1) - 2;      // first of the 12 source lines (pre-clamp)

  const int parity = lo & 1;
  const int xr     = parity * 3;            // XOR constant for phase reversal
  const int s_rel  = (lo >> 1) + parity;    // first nonzero tap (relative to base_src)

  // Per-lane column pointers; all in-kernel offsets are 32-bit (v_mul_lo_u32).
  const float* colp = in  + (long long)o * NIN * inner        + (x0 + lo);
  float*       ocol = out + (long long)o * (2 * NIN) * inner  + (x0 + lo);

  // Build all A weights and issue all 6 B loads before any WMMA.
  v2f a[3], b[3];
  #pragma unroll
  for (int g = 0; g < 3; ++g) {
    const int jx = 4 * g + 2 * hi;          // K indices held by this lane
    const int jy = jx + 1;
    const int tx = jx - s_rel;
    const int ty = jy - s_rel;
    // numerator byte from packed table; zeroed outside window via sign-mask AND
    unsigned nx = (0x1BEB7901u >> (8u * ((unsigned)(tx ^ xr) & 3u))) & 0xFFu;
    unsigned ny = (0x1BEB7901u >> (8u * ((unsigned)(ty ^ xr) & 3u))) & 0xFFu;
    nx &= (unsigned)(-(int)((unsigned)tx < 4u));
    ny &= (unsigned)(-(int)((unsigned)ty < 4u));
    a[g].x = (float)nx * (1.0f / 384.0f);
    a[g].y = (float)ny * (1.0f / 384.0f);
    b[g].x = colp[iclamp95(base_src + jx) * inner];
    b[g].y = colp[iclamp95(base_src + jy) * inner];
  }

  v8f acc = {0.f, 0.f, 0.f, 0.f, 0.f, 0.f, 0.f, 0.f};
  #pragma unroll
  for (int g = 0; g < 3; ++g)
    acc = __builtin_amdgcn_wmma_f32_16x16x4_f32(
        /*neg_a=*/false, a[g], /*neg_b=*/false, b[g],
        /*c_mod=*/(short)0, acc, /*reuse_a=*/false, /*reuse_b=*/false);

  #pragma unroll
  for (int vv = 0; vv < 8; ++vv)
    ocol[(zo0 + vv + 8 * hi) * inner] = acc[vv];
}

// ---------------------------------------------------------------------------
// x2 upsample along contiguous x: (R,96) -> (R,192). Coalesced NT stores
// (output is write-once, keep L2 for the staged intermediates).
// ---------------------------------------------------------------------------
__global__ __launch_bounds__(256)
void ups_x_kernel(const float* __restrict__ in, float* __restrict__ out, int total) {
  int tid = blockIdx.x * 256 + threadIdx.x;
  if (tid >= total) return;
  int xo = tid % 192;
  int r  = tid / 192;
  int parity = xo & 1;
  int s = (xo >> 1) + parity - 2;
  const float* row = in + (long long)r * 96;
  float acc = 0.f;
  #pragma unroll
  for (int c = 0; c < 4; ++c)
    acc = fmaf(wtap(parity, c), row[iclamp95(s + c)], acc);
  __builtin_nontemporal_store(acc, &out[tid]);
}

// ---------------------------------------------------------------------------
// Fallback: fused 3D 4x4x4 upsample directly from the 21 MB prefiltered buffer
// (used only when workspace is too small for staged intermediates).
// ---------------------------------------------------------------------------
__global__ __launch_bounds__(256)
void ups_fused_kernel(const float* __restrict__ A, float* __restrict__ out, int total) {
  int tid = blockIdx.x * 256 + threadIdx.x;
  if (tid >= total) return;
  int xo = tid % 192; int t1 = tid / 192;
  int yo = t1 % 192;  int t2 = t1 / 192;
  int zo = t2 % 192;  int bc = t2 / 192;
  int px = xo & 1, py = yo & 1, pz = zo & 1;
  int sx = (xo >> 1) + px - 2;
  int sy = (yo >> 1) + py - 2;
  int sz = (zo >> 1) + pz - 2;
  const float* base = A + (long long)bc * (96 * 96 * 96);
  float acc = 0.f;
  #pragma unroll
  for (int cz = 0; cz < 4; ++cz) {
    const float wz = wtap(pz, cz);
    const float* zp = base + (long long)iclamp95(sz + cz) * (96 * 96);
    #pragma unroll
    for (int cy = 0; cy < 4; ++cy) {
      const float wzy = wz * wtap(py, cy);
      const float* rp = zp + iclamp95(sy + cy) * 96;
      float rs = 0.f;
      #pragma unroll
      for (int cx = 0; cx < 4; ++cx)
        rs = fmaf(wtap(px, cx), rp[iclamp95(sx + cx)], rs);
      acc = fmaf(wzy, rs, acc);
    }
  }
  __builtin_nontemporal_store(acc, &out[tid]);
}

} // namespace

extern "C" void kernel_launch(void* const* d_in, const int* in_sizes, int n_in,
                              void* d_out, int out_size, void* d_ws, size_t ws_size,
                              hipStream_t stream) {
  const float* vol = (const float*)d_in[0];   // (2,3,96,96,96) f32
  float* out = (float*)d_out;                 // (2,3,192,192,192) f32
  float* A   = (float*)d_ws;                  // prefiltered coefficients, 21.2 MB

  const size_t A_elems = (size_t)6 * 96 * 96 * 96;     //  5,308,416
  const size_t C_elems = (size_t)6 * 192 * 96 * 96;    // 10,616,832
  const size_t D_elems = (size_t)6 * 192 * 192 * 96;   // 21,233,664
  const int    OUT_TOT = 42467328;                     // 6*192^3

  // --- prefilter (in-place on A after the first copy-through pass) ---
  pf_x_kernel<<<432, 128, 0, stream>>>(vol, A);                 // axis 4 (x), LDS transpose
  pf_strided_kernel<<<216, 256, 0, stream>>>(A, 96,   55296);   // axis 3 (y)
  pf_strided_kernel<<<216, 256, 0, stream>>>(A, 9216, 55296);   // axis 2 (z)

  if (ws_size >= (A_elems + C_elems + D_elems) * sizeof(float)) {
    // Staged separable upsampling; intermediates C (42.5 MB) and D (85 MB) stay in L2.
    float* C = A + A_elems;
    float* D = C + C_elems;
    // z: (6,96,9216) -> (6,192,9216): waves = 6*12*576 = 41472 -> 5184 blocks of 8 waves
    ups_axis_wmma_kernel<<<5184, 256, 0, stream>>>(A, C, 9216, 576);
    // y: (1152,96,96) -> (1152,192,96): waves = 1152*12*6 = 82944 -> 10368 blocks
    ups_axis_wmma_kernel<<<10368, 256, 0, stream>>>(C, D, 96, 6);
    // x: (221184,96) -> (221184,192)
    ups_x_kernel<<<165888, 256, 0, stream>>>(D, out, OUT_TOT);
  } else {
    // Workspace-lean fallback: fused 4x4x4 gather from A.
    ups_fused_kernel<<<165888, 256, 0, stream>>>(A, out, OUT_TOT);
  }
}